// CrossAttention_62998580298208
// MI455X (gfx1250) — compile-verified
//
#include <hip/hip_runtime.h>
#include <math.h>

// ---------------------------------------------------------------------------
// CrossAttention, algebraically collapsed (single query token):
//   scores = x2 @ u ,  u = (1/8) * We2 @ (Wk per-head-contracted with q)
//   y      = softmax(scores) @ x3  (per head)
//   out    = ((y @ We2) + be2-term) @ Wv  -> @ Wo + bo
// Roofline: traffic = x2+x3 = 268 MB -> ~11.5us @ 23.3 TB/s; FLOPs ~2.5 G.
// Dominant passes use V_WMMA_F32_16X16X4_F32 (exact fp32) + global_prefetch
// for the HBM streams.
// ---------------------------------------------------------------------------

typedef __attribute__((ext_vector_type(2))) float v2f;
typedef __attribute__((ext_vector_type(8))) float v8f;

#define B_  16
#define S_  4096
#define DIN 512
#define D_  1024
#define H_  16
#define HD_ 64

// ---------------- small GEMV-style kernels (L2-resident weights) -----------

// x1e[b,j] = be1[j] + sum_c x1[b,c] * We1[c,j]
__global__ void k0_x1e(const float* __restrict__ x1, const float* __restrict__ We1,
                       const float* __restrict__ be1, float* __restrict__ x1e) {
  int idx = blockIdx.x * blockDim.x + threadIdx.x;     // 16*1024
  int b = idx >> 10, j = idx & 1023;
  float acc = be1[j];
  const float* xr = x1 + b * DIN;
  for (int c = 0; c < DIN; ++c) acc = fmaf(xr[c], We1[c * D_ + j], acc);
  x1e[idx] = acc;
}

// qv[b,j] = bq[j] + sum_m x1e[b,m] * Wq[m,j]
__global__ void k1_q(const float* __restrict__ x1e, const float* __restrict__ Wq,
                     const float* __restrict__ bq, float* __restrict__ qv) {
  int idx = blockIdx.x * blockDim.x + threadIdx.x;     // 16*1024
  int b = idx >> 10, j = idx & 1023;
  float acc = bq[j];
  const float* xr = x1e + b * D_;
  for (int m = 0; m < D_; ++m) acc = fmaf(xr[m], Wq[m * D_ + j], acc);
  qv[idx] = acc;
}

// th[b,h,m] = sum_{d<64} Wk[m, h*64+d] * qv[b, h*64+d]
__global__ void k2_th(const float* __restrict__ Wk, const float* __restrict__ qv,
                      float* __restrict__ th) {
  int idx = blockIdx.x * blockDim.x + threadIdx.x;     // 16*16*1024
  int b = idx >> 14, h = (idx >> 10) & 15, m = idx & 1023;
  const float* wr = Wk + m * D_ + h * HD_;
  const float* qr = qv + b * D_ + h * HD_;
  float acc = 0.f;
  for (int d = 0; d < HD_; ++d) acc = fmaf(wr[d], qr[d], acc);
  th[idx] = acc;
}

// u[b,h,c] = (1/8) * sum_m We2[c,m] * th[b,h,m]   (softmax scale folded in)
__global__ void k3_u(const float* __restrict__ We2, const float* __restrict__ th,
                     float* __restrict__ u) {
  int idx = blockIdx.x * blockDim.x + threadIdx.x;     // 16*16*512
  int b = idx >> 13, h = (idx >> 9) & 15, c = idx & 511;
  const float* wr = We2 + c * D_;
  const float* tr = th + (b * H_ + h) * D_;
  float acc = 0.f;
  for (int m = 0; m < D_; ++m) acc = fmaf(wr[m], tr[m], acc);
  u[idx] = acc * 0.125f;
}

// ---------------- Pass A: scores[b,h,s] = X2_b[16xK] @ U_b^T  (WMMA f32) ---
// One wave per (b, s-tile of 16). K = 512, step 4 -> 128 v_wmma per wave.
__global__ __launch_bounds__(32)
void k4_scores(const float* __restrict__ x2, const float* __restrict__ u,
               float* __restrict__ scores) {
  int b = blockIdx.x >> 8;
  int s0 = (blockIdx.x & 255) << 4;
  int lane = threadIdx.x;
  int hi = lane >> 4;        // K half-pair select
  int r  = lane & 15;        // A: row M ; B: col N (head)
  const float* arow = x2 + ((size_t)b * S_ + (s0 + r)) * DIN + hi * 2;
  const float* brow = u + ((size_t)b * H_ + r) * DIN + hi * 2;
  v8f acc = {};
  for (int k = 0; k < DIN; k += 4) {
    if ((k & 15) == 0)  // uniform; folds away under unroll -> global_prefetch
      __builtin_prefetch((const void*)(arow + k + 256), 0, 1);
    v2f a = *(const v2f*)(arow + k);
    v2f bb = *(const v2f*)(brow + k);
    acc = __builtin_amdgcn_wmma_f32_16x16x4_f32(false, a, false, bb,
                                                (short)0, acc, false, false);
  }
  // D[M][N]: lane holds N=r, M = vr + hi*8 -> scores[b][h=r][s0 + M]
  float* sc = scores + ((size_t)(b * H_ + r)) * S_ + s0 + hi * 8;
#pragma unroll
  for (int vr = 0; vr < 8; ++vr) sc[vr] = acc[vr];
}

// ---------------- softmax over s (in-place exp, store row sums) ------------
__global__ __launch_bounds__(256)
void k5_softmax(float* __restrict__ scores, float* __restrict__ sums) {
  __shared__ float red[256];
  int tid = threadIdx.x;
  float* row = scores + (size_t)blockIdx.x * S_;
  float mx = -INFINITY;
  for (int s = tid; s < S_; s += 256) mx = fmaxf(mx, row[s]);
  red[tid] = mx; __syncthreads();
  for (int off = 128; off; off >>= 1) {
    if (tid < off) red[tid] = fmaxf(red[tid], red[tid + off]);
    __syncthreads();
  }
  mx = red[0]; __syncthreads();
  float sum = 0.f;
  for (int s = tid; s < S_; s += 256) {
    float e = __expf(row[s] - mx);
    row[s] = e;
    sum += e;
  }
  red[tid] = sum; __syncthreads();
  for (int off = 128; off; off >>= 1) {
    if (tid < off) red[tid] += red[tid + off];
    __syncthreads();
  }
  if (tid == 0) sums[blockIdx.x] = red[0];
}

// ---------------- Pass B: y_part[kp,b,h,c] = P[16xK] @ X3[Kx16]  (WMMA) ----
// grid = b(16) x ctile(32) x kpart(4); deterministic partial sums (no atomics)
__global__ __launch_bounds__(32)
void k6_wsum(const float* __restrict__ p, const float* __restrict__ x3,
             float* __restrict__ y_part) {
  int idx = blockIdx.x;
  int kp = idx & 3, ct = (idx >> 2) & 31, b = idx >> 7;
  int c0 = ct << 4;
  int lane = threadIdx.x;
  int hi = lane >> 4;
  int r  = lane & 15;
  const float* pb  = p + ((size_t)b * H_ + r) * S_ + hi * 2;   // A: M=r (head)
  const float* x3b = x3 + (size_t)b * S_ * DIN + c0 + r;       // B: N=r (chan)
  int kbeg = kp * (S_ / 4), kend = kbeg + (S_ / 4);
  v8f acc = {};
  for (int k = kbeg; k < kend; k += 4) {
    if ((k & 15) == 0)  // prefetch x3 stream 16 s-rows ahead
      __builtin_prefetch((const void*)(x3b + (size_t)(k + 16 + hi * 2) * DIN), 0, 1);
    v2f a = *(const v2f*)(pb + k);
    v2f bb;
    bb[0] = x3b[(size_t)(k + hi * 2) * DIN];
    bb[1] = x3b[(size_t)(k + hi * 2 + 1) * DIN];
    acc = __builtin_amdgcn_wmma_f32_16x16x4_f32(false, a, false, bb,
                                                (short)0, acc, false, false);
  }
  // D[M][N]: head M = vr + hi*8, channel N = r
  float* yo = y_part + (((size_t)(kp * B_ + b) * H_ + hi * 8) * DIN) + c0 + r;
#pragma unroll
  for (int vr = 0; vr < 8; ++vr) yo[(size_t)vr * DIN] = acc[vr];
}

// ybar[b,h,c] = (sum_kp y_part) / sums[b,h]
__global__ void k6b_combine(const float* __restrict__ y_part,
                            const float* __restrict__ sums,
                            float* __restrict__ ybar) {
  int idx = blockIdx.x * blockDim.x + threadIdx.x;     // 16*16*512
  int bh = idx >> 9;
  float v = y_part[idx] + y_part[idx + B_ * H_ * DIN] +
            y_part[idx + 2 * B_ * H_ * DIN] + y_part[idx + 3 * B_ * H_ * DIN];
  ybar[idx] = v / sums[bh];
}

// z[b,h,m] = sum_c ybar[b,h,c] * We2[c,m]
__global__ void k7_z(const float* __restrict__ ybar, const float* __restrict__ We2,
                     float* __restrict__ z) {
  int idx = blockIdx.x * blockDim.x + threadIdx.x;     // 16*16*1024
  int bh = idx >> 10, m = idx & 1023;
  const float* yr = ybar + bh * DIN;
  float acc = 0.f;
  for (int c = 0; c < DIN; ++c) acc = fmaf(yr[c], We2[c * D_ + m], acc);
  z[idx] = acc;
}

// av[b,j] = bv[j] + sum_m (z[b,h,m] + be2[m]) * Wv[m,j] , h = j/64
__global__ void k8_av(const float* __restrict__ z, const float* __restrict__ be2,
                      const float* __restrict__ Wv, const float* __restrict__ bv,
                      float* __restrict__ av) {
  int idx = blockIdx.x * blockDim.x + threadIdx.x;     // 16*1024
  int b = idx >> 10, j = idx & 1023, h = j >> 6;
  const float* zr = z + (b * H_ + h) * D_;
  float acc = bv[j];
  for (int m = 0; m < D_; ++m) acc = fmaf(zr[m] + be2[m], Wv[m * D_ + j], acc);
  av[idx] = acc;
}

// out[b,j] = bo[j] + sum_i av[b,i] * Wo[i,j]
__global__ void k9_out(const float* __restrict__ av, const float* __restrict__ Wo,
                       const float* __restrict__ bo, float* __restrict__ out) {
  int idx = blockIdx.x * blockDim.x + threadIdx.x;     // 16*1024
  int b = idx >> 10, j = idx & 1023;
  const float* ar = av + b * D_;
  float acc = bo[j];
  for (int i = 0; i < D_; ++i) acc = fmaf(ar[i], Wo[i * D_ + j], acc);
  out[idx] = acc;
}

// ---------------------------------------------------------------------------
extern "C" void kernel_launch(void* const* d_in, const int* in_sizes, int n_in,
                              void* d_out, int out_size, void* d_ws, size_t ws_size,
                              hipStream_t stream) {
  const float* x1  = (const float*)d_in[0];
  const float* x2  = (const float*)d_in[1];
  const float* x3  = (const float*)d_in[2];
  const float* We1 = (const float*)d_in[3];
  const float* be1 = (const float*)d_in[4];
  const float* We2 = (const float*)d_in[5];
  const float* be2 = (const float*)d_in[6];
  const float* Wq  = (const float*)d_in[7];
  const float* bq  = (const float*)d_in[8];
  const float* Wk  = (const float*)d_in[9];
  // const float* bk = (const float*)d_in[10];  // cancels in softmax
  const float* Wv  = (const float*)d_in[11];
  const float* bv  = (const float*)d_in[12];
  const float* Wo  = (const float*)d_in[13];
  const float* bo  = (const float*)d_in[14];
  float* out = (float*)d_out;

  float* ws = (float*)d_ws;
  float* x1e    = ws;                      // 16384
  float* qv     = x1e + 16384;             // 16384
  float* th     = qv + 16384;              // 262144
  float* u      = th + 262144;             // 131072
  float* scores = u + 131072;              // 1048576 (reused in-place as p)
  float* sums   = scores + 1048576;        // 256
  float* y_part = sums + 256;              // 524288
  float* ybar   = y_part + 524288;         // 131072
  float* z      = ybar + 131072;           // 262144
  float* av     = z + 262144;              // 16384   (total ~9.2 MB)

  k0_x1e<<<64, 256, 0, stream>>>(x1, We1, be1, x1e);
  k1_q<<<64, 256, 0, stream>>>(x1e, Wq, bq, qv);
  k2_th<<<1024, 256, 0, stream>>>(Wk, qv, th);
  k3_u<<<512, 256, 0, stream>>>(We2, th, u);
  k4_scores<<<B_ * (S_ / 16), 32, 0, stream>>>(x2, u, scores);
  k5_softmax<<<B_ * H_, 256, 0, stream>>>(scores, sums);
  k6_wsum<<<B_ * 32 * 4, 32, 0, stream>>>(scores, x3, y_part);
  k6b_combine<<<512, 256, 0, stream>>>(y_part, sums, ybar);
  k7_z<<<1024, 256, 0, stream>>>(ybar, We2, z);
  k8_av<<<64, 256, 0, stream>>>(z, be2, Wv, bv, av);
  k9_out<<<64, 256, 0, stream>>>(av, Wo, bo, out);
}